// HomomorphicVocoder_65704409694586
// MI455X (gfx1250) — compile-verified
//
#include <hip/hip_runtime.h>
#include <math.h>

// ---------------- problem constants ----------------
#define N_FFT 1920
#define HOP   480
#define SRATE 48000
#define NB    16
#define NTIME 600
#define TF    601                    // frames per batch (T+1)
#define NT    (NB*TF)                // 9616 frame-columns, multiple of 16
#define NF    961                    // onesided bins
#define FP    976                    // padded bins (61*16)
#define M2    (2*FP)                 // 1952 stacked Re/Im rows
#define LSIG  288000                 // NTIME*HOP
#define OLEN  (N_FFT + HOP*(TF-1))   // 289920
#define KREV  8192
#define NCH   150                    // scan chunks per batch
#define CHN   1920                   // chunk length (NCH*CHN==LSIG)
#define KC    16                     // GEMM K-chunk (all K are multiples of 16)

typedef float v2f __attribute__((ext_vector_type(2)));
typedef float v8f __attribute__((ext_vector_type(8)));

static __device__ __forceinline__ float hannw(int k) {
  return 0.5f * (1.0f - cosf(6.28318530717958647692f * (float)k / (float)N_FFT));
}

// linear upsample of f0 by HOP (align_corners=False semantics)
static __device__ __forceinline__ double f0_up_val(const float* __restrict__ f0, int b, int i) {
  double pos = ((double)i + 0.5) / (double)HOP - 0.5;
  if (pos < 0.0) pos = 0.0;
  if (pos > (double)(NTIME - 1)) pos = (double)(NTIME - 1);
  int i0 = (int)pos;
  int i1 = i0 + 1; if (i1 > NTIME - 1) i1 = NTIME - 1;
  double w = pos - (double)i0;
  const float* row = f0 + (size_t)b * NTIME;
  return (double)row[i0] * (1.0 - w) + (double)row[i1] * w;
}

// ---------------- trig matrix builders ----------------
// A_stft [M2 x N_FFT]: rows f<961: cos(2pi f n/N); rows 976+f: -sin(2pi f n/N); pad rows 0
__global__ void hv_build_A(float* __restrict__ A) {
  int idx = blockIdx.x * blockDim.x + threadIdx.x;
  if (idx >= M2 * N_FFT) return;
  int m = idx / N_FFT, n = idx % N_FFT;
  float v = 0.f;
  if (m < NF) {
    int r = (int)(((long long)m * n) % N_FFT);
    v = cosf(6.28318530717958647692f * (float)r / (float)N_FFT);
  } else if (m >= FP && m < FP + NF) {
    int f = m - FP;
    int r = (int)(((long long)f * n) % N_FFT);
    v = -sinf(6.28318530717958647692f * (float)r / (float)N_FFT);
  }
  A[idx] = v;
}

// CC_lift [N_FFT x FP]: lifter[n] * (c_f/N) * cos(2pi f n/N)
__global__ void hv_build_CC(float* __restrict__ CC) {
  int idx = blockIdx.x * blockDim.x + threadIdx.x;
  if (idx >= N_FFT * FP) return;
  int n = idx / FP, fc = idx % FP;
  float v = 0.f;
  if (fc < NF) {
    float wl = (n == 0 || n == N_FFT / 2) ? 1.f : (n < N_FFT / 2 ? 2.f : 0.f);
    if (wl != 0.f) {
      float cf = (fc == 0 || fc == NF - 1) ? 1.f : 2.f;
      int r = (int)(((long long)n * fc) % N_FFT);
      v = wl * (cf / (float)N_FFT) * cosf(6.28318530717958647692f * (float)r / (float)N_FFT);
    }
  }
  CC[idx] = v;
}

// IS_cat [N_FFT x M2]: cols f<961: (c_f/N) cos ; cols 976+f: -(c_f/N) sin ; pad cols 0
__global__ void hv_build_IS(float* __restrict__ IS) {
  int idx = blockIdx.x * blockDim.x + threadIdx.x;
  if (idx >= N_FFT * M2) return;
  int n = idx / M2, q = idx % M2;
  float v = 0.f;
  if (q < NF) {
    float cf = (q == 0 || q == NF - 1) ? 1.f : 2.f;
    int r = (int)(((long long)n * q) % N_FFT);
    v = (cf / (float)N_FFT) * cosf(6.28318530717958647692f * (float)r / (float)N_FFT);
  } else if (q >= FP && q < FP + NF) {
    int f = q - FP;
    float cf = (f == 0 || f == NF - 1) ? 1.f : 2.f;
    int r = (int)(((long long)n * f) % N_FFT);
    v = -(cf / (float)N_FFT) * sinf(6.28318530717958647692f * (float)r / (float)N_FFT);
  }
  IS[idx] = v;
}

// ---------------- impulse train (3-phase chunked scan, f64) ----------------
__global__ void hv_scan_chunks(const float* __restrict__ f0, double* __restrict__ csum) {
  int idx = blockIdx.x * blockDim.x + threadIdx.x;
  if (idx >= NB * NCH) return;
  int b = idx / NCH, c = idx % NCH;
  double s = 0.0;
  int base = c * CHN;
  for (int j = 0; j < CHN; ++j) s += f0_up_val(f0, b, base + j) / (double)SRATE;
  csum[idx] = s;
}

__global__ void hv_scan_offsets(double* __restrict__ csum) {
  int b = threadIdx.x;
  if (b >= NB) return;
  double run = 0.0;
  for (int c = 0; c < NCH; ++c) {
    double t = csum[b * NCH + c];
    csum[b * NCH + c] = run;
    run += t;
  }
}

__global__ void hv_impulse(const float* __restrict__ f0, const double* __restrict__ csum,
                           float* __restrict__ imp) {
  int idx = blockIdx.x * blockDim.x + threadIdx.x;
  if (idx >= NB * NCH) return;
  int b = idx / NCH, c = idx % NCH;
  double phase = csum[idx];
  double prev = (c == 0) ? 0.0 : (phase - floor(phase));
  int base = c * CHN;
  for (int j = 0; j < CHN; ++j) {
    int i = base + j;
    double v = f0_up_val(f0, b, i);
    phase += v / (double)SRATE;
    double saw = phase - floor(phase);
    float o = 0.f;
    if (saw < prev) {
      double vm = (v < 20.0) ? 20.0 : v;
      o = (float)sqrt((double)SRATE / vm);   // impulse * rsqrt(max(f0,20))*sqrt(SR)
    }
    imp[(size_t)b * LSIG + i] = o;
    prev = saw;
  }
}

// ---------------- frame building (mixed excitation, windowed, reflect-pad) ----------------
__global__ void hv_frames(const float* __restrict__ f0, const float* __restrict__ per,
                          const float* __restrict__ noi, const float* __restrict__ imp,
                          float* __restrict__ FR) {
  int idx = blockIdx.x * blockDim.x + threadIdx.x;
  if (idx >= N_FFT * NT) return;
  int col = idx % NT, k = idx / NT;
  int b = col / TF, t = col % TF;
  float pt = 0.f;
  if (t > 0) {
    float f0v = f0[b * NTIME + t - 1];
    if (f0v > 20.f) pt = per[b * NTIME + t - 1];
  }
  int s = t * HOP + k - N_FFT / 2;
  if (s < 0) s = -s;
  if (s >= LSIG) s = 2 * LSIG - 2 - s;
  size_t si = (size_t)b * LSIG + s;
  float x = (1.f - pt) * noi[si] + pt * imp[si];
  FR[idx] = hannw(k) * x;
}

// ---------------- log envelope (left-pad one frame of zeros) ----------------
__global__ void hv_logenv(const float* __restrict__ env, float* __restrict__ LE) {
  int idx = blockIdx.x * blockDim.x + threadIdx.x;
  if (idx >= FP * NT) return;
  int col = idx % NT, f = idx / NT;
  float v = 0.f;
  if (f < NF) {
    int b = col / TF, t = col % TF;
    float e = (t == 0) ? 0.f : env[((size_t)b * NF + f) * NTIME + (t - 1)];
    v = logf(fmaxf(e, 1e-5f));
  }
  LE[idx] = v;
}

// ---------------- WMMA f32 GEMM: C[M,N] = A[M,K] * B[K,N], row-major ----------------
// 256 threads = 8 waves; block tile 128x64; wave tile 16x64 (4 accumulators).
// B chunks (KC x 64) staged via LDS, pair-interleaved so each lane's K-pair is one
// ds_load_b64; double-buffered. Both A fragments and B chunks are prefetched one
// chunk ahead into registers so the 16 WMMAs per chunk wait only on DS counters.
// Requires K % KC == 0, M % 16 == 0; N tail via clamped loads + guarded stores.
__global__ __launch_bounds__(256)
void hv_gemm_wmma(const float* __restrict__ A, const float* __restrict__ B,
                  float* __restrict__ C, int M, int K, int N) {
  __shared__ float lb[2][(KC / 2) * 128];   // 2 x 4KB

  const int tid  = threadIdx.x;
  const int wave = tid >> 5;
  const int lane = tid & 31;
  const int half = lane >> 4;               // which K-pair of the 4 this lane holds
  const int l    = lane & 15;
  const int m0   = blockIdx.y * 128 + wave * 16;
  const int n0   = blockIdx.x * 64;
  const bool active = (m0 < M);             // wave-uniform

  // B staging mapping: thread -> pair row kp (0..7), col pair c0
  const int kp = tid >> 5;                  // 8 pair-rows per chunk
  const int c0 = (tid & 31) * 2;            // 2 columns per thread
  const int gc0 = n0 + c0;
  const int gcl0 = (gc0     < N) ? gc0     : N - 1;
  const int gcl1 = (gc0 + 1 < N) ? gc0 + 1 : N - 1;

  const float* pB0 = B + (size_t)(2 * kp) * N;      // even row of this thread's pair
  const float* pB1 = pB0 + N;                        // odd row
  const size_t bstep = (size_t)KC * N;

  const int nch = K / KC;

  // ---- stage chunk 0 (B to LDS, A fragments to registers) ----
  float x0 = pB0[gcl0], x1 = pB0[gcl1];
  float y0 = pB1[gcl0], y1 = pB1[gcl1];
  {
    float* st = &lb[0][kp * 128 + c0 * 2];
    st[0] = x0; st[1] = y0; st[2] = x1; st[3] = y1;   // pairs {B[2p][c],B[2p+1][c]}
  }
  pB0 += bstep; pB1 += bstep;

  const float* arp = active ? (A + (size_t)(m0 + l) * K + 2 * half)
                            : (A + 2 * half);         // harmless in-range dummy
  v2f av0, av1, av2, av3;                             // A fragments, current chunk
  v2f nv0 = {}, nv1 = {}, nv2 = {}, nv3 = {};         // A fragments, next chunk
  av0 = *(const v2f*)(arp + 0);
  av1 = *(const v2f*)(arp + 4);
  av2 = *(const v2f*)(arp + 8);
  av3 = *(const v2f*)(arp + 12);
  arp += KC;
  __syncthreads();

  v8f acc[4] = {v8f{}, v8f{}, v8f{}, v8f{}};

  for (int ch = 0; ch < nch; ++ch) {
    const int cur = ch & 1;
    const bool more = (ch + 1 < nch);
    // prefetch next chunk (B -> regs, A -> regs); overlaps the WMMAs below
    if (more) {
      x0 = pB0[gcl0]; x1 = pB0[gcl1];
      y0 = pB1[gcl0]; y1 = pB1[gcl1];
      pB0 += bstep; pB1 += bstep;
      nv0 = *(const v2f*)(arp + 0);
      nv1 = *(const v2f*)(arp + 4);
      nv2 = *(const v2f*)(arp + 8);
      nv3 = *(const v2f*)(arp + 12);
      arp += KC;
    }
    if (active) {
#pragma unroll
      for (int kk = 0; kk < KC; kk += 4) {
        v2f a = (kk == 0) ? av0 : (kk == 4) ? av1 : (kk == 8) ? av2 : av3;
        const float* ls = &lb[cur][(kk / 2 + half) * 128];
#pragma unroll
        for (int j = 0; j < 4; ++j) {
          v2f bv = *(const v2f*)(ls + (16 * j + l) * 2);   // one ds_load_b64
          acc[j] = __builtin_amdgcn_wmma_f32_16x16x4_f32(false, a, false, bv,
                                                         (short)0, acc[j], false, false);
        }
      }
    }
    __syncthreads();
    if (more) {
      float* st = &lb[cur ^ 1][kp * 128 + c0 * 2];
      st[0] = x0; st[1] = y0; st[2] = x1; st[3] = y1;
      av0 = nv0; av1 = nv1; av2 = nv2; av3 = nv3;
    }
    __syncthreads();
  }

  if (active) {
#pragma unroll
    for (int j = 0; j < 4; ++j) {
      int col = n0 + 16 * j + l;
      if (col < N) {
        float* cp = C + (size_t)(m0 + half * 8) * N + col;  // C layout: VGPR r -> row r (+8 hi half)
#pragma unroll
        for (int r = 0; r < 8; ++r) cp[(size_t)r * N] = acc[j][r];
      }
    }
  }
}

// ---------------- spectral combine: V = exc * exp(MP) (complex), in-place into SX ----------------
__global__ void hv_combine(const float* __restrict__ MP, float* __restrict__ SX) {
  int idx = blockIdx.x * blockDim.x + threadIdx.x;
  if (idx >= NF * NT) return;
  int f = idx / NT, col = idx % NT;
  size_t ir = (size_t)f * NT + col;
  size_t ii = (size_t)(f + FP) * NT + col;
  float mr = MP[ir], mi = MP[ii];
  float ex = expf(mr);
  float er = ex * cosf(mi), ei = ex * sinf(mi);
  float sr = SX[ir], si = SX[ii];
  SX[ir] = sr * er - si * ei;
  SX[ii] = sr * ei + si * er;
}

__global__ void hv_zero(float* __restrict__ p, int n) {
  int idx = blockIdx.x * blockDim.x + threadIdx.x;
  if (idx < n) p[idx] = 0.f;
}

// ---------------- windowed overlap-add ----------------
__global__ void hv_ola(const float* __restrict__ Y, float* __restrict__ voi) {
  int idx = blockIdx.x * blockDim.x + threadIdx.x;
  if (idx >= N_FFT * NT) return;
  int col = idx % NT, n = idx / NT;
  int b = col / TF, t = col % TF;
  float v = Y[idx] * hannw(n);
  atomicAdd(&voi[(size_t)b * OLEN + t * HOP + n], v);
}

// ---------------- normalize by window-square sum, crop center pad ----------------
__global__ void hv_normcrop(const float* __restrict__ voi, float* __restrict__ vf) {
  int idx = blockIdx.x * blockDim.x + threadIdx.x;
  if (idx >= NB * LSIG) return;
  int b = idx / LSIG, i = idx % LSIG;
  int j = N_FFT / 2 + i;
  int tlo = (j - (N_FFT - 1) + (HOP - 1)) / HOP;
  if (tlo < 0) tlo = 0;
  int thi = j / HOP;
  if (thi > TF - 1) thi = TF - 1;
  float ws = 0.f;
  for (int t = tlo; t <= thi; ++t) {
    float w = hannw(j - HOP * t);
    ws += w * w;
  }
  vf[idx] = voi[(size_t)b * OLEN + j] / fmaxf(ws, 1e-11f);
}

// ---------------- causal reverb convolution (LDS-tiled time domain) ----------------
__global__ __launch_bounds__(256)
void hv_conv(const float* __restrict__ vf, const float* __restrict__ rev,
             float* __restrict__ out) {
  __shared__ float sr[256];
  __shared__ float sx[512];
  int b = blockIdx.y;
  int t0 = blockIdx.x * 256;
  int tid = threadIdx.x;
  float acc = 0.f;
  for (int k0 = 0; k0 < KREV; k0 += 256) {
    __syncthreads();
    sr[tid] = rev[(size_t)b * KREV + k0 + tid];
    int base = t0 - k0 - 255;
#pragma unroll
    for (int i = tid; i < 512; i += 256) {
      int s = base + i;
      sx[i] = (s >= 0 && s < LSIG) ? vf[(size_t)b * LSIG + s] : 0.f;
    }
    __syncthreads();
#pragma unroll 8
    for (int kk = 0; kk < 256; ++kk) acc += sr[kk] * sx[tid + 255 - kk];
  }
  out[(size_t)b * LSIG + t0 + tid] = acc;
}

// ---------------- orchestration ----------------
static inline unsigned cdiv_u(size_t a, unsigned b) { return (unsigned)((a + b - 1) / b); }

extern "C" void kernel_launch(void* const* d_in, const int* in_sizes, int n_in,
                              void* d_out, int out_size, void* d_ws, size_t ws_size,
                              hipStream_t stream) {
  (void)in_sizes; (void)n_in; (void)out_size; (void)ws_size;
  const float* f0  = (const float*)d_in[0];   // [16,600]
  const float* per = (const float*)d_in[1];   // [16,1,600]
  const float* env = (const float*)d_in[2];   // [16,961,600]
  const float* rev = (const float*)d_in[3];   // [16,8192]
  const float* noi = (const float*)d_in[4];   // [16,288000]
  float* out = (float*)d_out;                 // [16,288000]

  float* w = (float*)d_ws;
  size_t oA  = 0;
  size_t oCC = oA  + (size_t)M2 * N_FFT;      // A_stft  1952x1920
  size_t oIS = oCC + (size_t)N_FFT * FP;      // CC_lift 1920x976
  size_t oIM = oIS + (size_t)N_FFT * M2;      // IS_cat  1920x1952
  size_t oFR = oIM + (size_t)NB * LSIG;       // imp / later vf
  size_t oSX = oFR + (size_t)N_FFT * NT;      // FrameExc / cep / Yframes
  size_t oLE = oSX + (size_t)M2 * NT;         // S_exc / V (in-place)
  size_t oMP = oLE + (size_t)FP * NT;         // LogEnv
  size_t oVO = oMP + (size_t)M2 * NT;         // MP
  size_t oCS = oVO + (size_t)NB * OLEN;       // voi (OLA accumulator)
  float* A  = w + oA;  float* CC = w + oCC; float* IS = w + oIS;
  float* IM = w + oIM; float* FR = w + oFR; float* SX = w + oSX;
  float* LE = w + oLE; float* MP = w + oMP; float* VO = w + oVO;
  double* CS = (double*)(w + oCS);

  const unsigned TB = 256;

  // trig matrices (in production built once; rebuilt per call for determinism)
  hv_build_A <<<cdiv_u((size_t)M2 * N_FFT, TB), TB, 0, stream>>>(A);
  hv_build_CC<<<cdiv_u((size_t)N_FFT * FP, TB), TB, 0, stream>>>(CC);
  hv_build_IS<<<cdiv_u((size_t)N_FFT * M2, TB), TB, 0, stream>>>(IS);

  // impulse train via chunked f64 scan
  hv_scan_chunks <<<cdiv_u(NB * NCH, TB), TB, 0, stream>>>(f0, CS);
  hv_scan_offsets<<<1, 32, 0, stream>>>(CS);
  hv_impulse     <<<cdiv_u(NB * NCH, TB), TB, 0, stream>>>(f0, CS, IM);

  // mixed excitation frames -> STFT GEMM (S_exc = A_stft x FrameExc)
  hv_frames<<<cdiv_u((size_t)N_FFT * NT, TB), TB, 0, stream>>>(f0, per, noi, IM, FR);
  hv_gemm_wmma<<<dim3(cdiv_u(NT, 64), cdiv_u(M2, 128)), 256, 0, stream>>>(A, FR, SX, M2, N_FFT, NT);

  // min-phase: cep = CC_lift x LogEnv ; MP = A_stft x cep
  hv_logenv<<<cdiv_u((size_t)FP * NT, TB), TB, 0, stream>>>(env, LE);
  hv_gemm_wmma<<<dim3(cdiv_u(NT, 64), cdiv_u(N_FFT, 128)), 256, 0, stream>>>(CC, LE, FR, N_FFT, FP, NT);
  hv_gemm_wmma<<<dim3(cdiv_u(NT, 64), cdiv_u(M2, 128)), 256, 0, stream>>>(A, FR, MP, M2, N_FFT, NT);

  // V = exc * exp(MP)  (in-place into SX)
  hv_combine<<<cdiv_u((size_t)NF * NT, TB), TB, 0, stream>>>(MP, SX);

  // istft: Yframes = IS_cat x V ; window + overlap-add ; normalize/crop
  hv_gemm_wmma<<<dim3(cdiv_u(NT, 64), cdiv_u(N_FFT, 128)), 256, 0, stream>>>(IS, SX, FR, N_FFT, M2, NT);
  hv_zero<<<cdiv_u((size_t)NB * OLEN, TB), TB, 0, stream>>>(VO, NB * OLEN);
  hv_ola <<<cdiv_u((size_t)N_FFT * NT, TB), TB, 0, stream>>>(FR, VO);
  hv_normcrop<<<cdiv_u((size_t)NB * LSIG, TB), TB, 0, stream>>>(VO, IM);  // vf reuses IM

  // reverb convolution
  hv_conv<<<dim3(LSIG / 256, NB), 256, 0, stream>>>(IM, rev, out);
}